// GAT_13657996001658
// MI455X (gfx1250) — compile-verified
//
#include <hip/hip_runtime.h>

#define N_NODES 100000
#define N_EDGES 1000000
#define D 64
#define L 3
#define RVQ 3
#define CODES 16
#define BN_EPS 1e-5f
#define COMMIT_W 0.25f
#define NTILES (N_NODES / 16)   // 6250 exactly

typedef float v2f __attribute__((ext_vector_type(2)));
typedef float v8f __attribute__((ext_vector_type(8)));

__device__ __forceinline__ v8f v8f_zero() {
    v8f z;
#pragma unroll
    for (int i = 0; i < 8; ++i) z[i] = 0.0f;
    return z;
}

// D = A(16x4 f32) * B(4x16 f32) + C(16x16 f32), full fp32 precision.
__device__ __forceinline__ v8f wmma_f32(v2f a, v2f b, v8f c) {
    return __builtin_amdgcn_wmma_f32_16x16x4_f32(false, a, false, b, (short)0, c, false, false);
}

// ---------------------------------------------------------------- utilities
__global__ void zero_kernel(float* __restrict__ p, int n) {
    int i = blockIdx.x * blockDim.x + threadIdx.x;
    if (i < n) p[i] = 0.0f;
}

// ------------------------------------------------- edge aggregation (L2-resident)
__global__ void edge_cnt_kernel(const int* __restrict__ ei, float* __restrict__ cnt) {
    int e = blockIdx.x * blockDim.x + threadIdx.x;
    if (e >= N_EDGES) return;
    atomicAdd(&cnt[ei[N_EDGES + e]], 1.0f);
}

__global__ void edge_agg_kernel(const int* __restrict__ ei, const float* __restrict__ h,
                                float* __restrict__ agg) {
    int idx = blockIdx.x * blockDim.x + threadIdx.x;
    int e = idx >> 4;                 // 16 chunks of 4 channels per edge
    if (e >= N_EDGES) return;
    int c4 = (idx & 15) << 2;
    int s = ei[e];
    int d = ei[N_EDGES + e];
    float4 v = *(const float4*)(h + (size_t)s * D + c4);
    float* dp = agg + (size_t)d * D + c4;
    atomicAdd(dp + 0, v.x);
    atomicAdd(dp + 1, v.y);
    atomicAdd(dp + 2, v.z);
    atomicAdd(dp + 3, v.w);
}

// --------------------------------------------------------- fused SAGE layer GEMM
// out = l2norm(mean@Wa + ba + h@Wr) + (h@Wl + bl), one wave per 16x64 tile.
__global__ __launch_bounds__(256) void sage_gemm_kernel(
    const float* __restrict__ h, const float* __restrict__ agg,
    const float* __restrict__ cnt,
    const float* __restrict__ Wa, const float* __restrict__ ba,
    const float* __restrict__ Wr,
    const float* __restrict__ Wl, const float* __restrict__ bl,
    float* __restrict__ out) {
    __shared__ float sW[3][D * D];    // transposed: sW[m][n*64+k] = W[k*64+n]
    int tid = threadIdx.x;
    for (int t = tid; t < D * D; t += 256) {
        int k = t >> 6, n = t & 63;
        sW[0][n * 64 + k] = Wa[t];
        sW[1][n * 64 + k] = Wr[t];
        sW[2][n * 64 + k] = Wl[t];
    }
    __syncthreads();

    int wave = tid >> 5, lane = tid & 31;
    int tile = blockIdx.x * 8 + wave;
    if (tile >= NTILES) return;

    int hi  = lane >> 4;              // K-pair selector
    int off = hi * 2;
    int col = lane & 15;
    int row = tile * 16 + col;        // A-operand row for this lane

    float scale = 1.0f / fmaxf(cnt[row], 1.0f);

    // Preload both A tiles (mean-agg and h) for this lane: 16 float2 each.
    v2f am[16], ah[16];
    const float* arow = agg + (size_t)row * D;
    const float* hrow = h + (size_t)row * D;
#pragma unroll
    for (int kk = 0; kk < 16; ++kk) {
        int k = kk * 4 + off;
        am[kk] = (*(const v2f*)(arow + k)) * scale;
        ah[kk] = *(const v2f*)(hrow + k);
    }

    v8f accS[4], accT[4];
#pragma unroll
    for (int nt = 0; nt < 4; ++nt) {
        v8f s = v8f_zero(), t = v8f_zero();
        const float* pa = sW[0] + (nt * 16 + col) * 64 + off;
        const float* pr = sW[1] + (nt * 16 + col) * 64 + off;
        const float* pl = sW[2] + (nt * 16 + col) * 64 + off;
#pragma unroll
        for (int kk = 0; kk < 16; ++kk) {
            v2f bwa = *(const v2f*)(pa + kk * 4);
            v2f bwr = *(const v2f*)(pr + kk * 4);
            v2f bwl = *(const v2f*)(pl + kk * 4);
            s = wmma_f32(am[kk], bwa, s);
            s = wmma_f32(ah[kk], bwr, s);
            t = wmma_f32(ah[kk], bwl, t);
        }
        accS[nt] = s;
        accT[nt] = t;
    }

    // Biases (per output channel) before the row-normalization.
#pragma unroll
    for (int nt = 0; nt < 4; ++nt) {
        float vba = ba[nt * 16 + col];
        float vbl = bl[nt * 16 + col];
#pragma unroll
        for (int v = 0; v < 8; ++v) {
            accS[nt][v] += vba;
            accT[nt][v] += vbl;
        }
    }

    // Row L2 norms of S: C layout row M = v + 8*hi, col = nt*16 + (lane&15).
    float sc[8];
#pragma unroll
    for (int v = 0; v < 8; ++v) {
        float p = 0.0f;
#pragma unroll
        for (int nt = 0; nt < 4; ++nt) p += accS[nt][v] * accS[nt][v];
        p += __shfl_xor(p, 1, 32);
        p += __shfl_xor(p, 2, 32);
        p += __shfl_xor(p, 4, 32);
        p += __shfl_xor(p, 8, 32);
        sc[v] = 1.0f / (sqrtf(p) + 1e-12f);
    }

#pragma unroll
    for (int nt = 0; nt < 4; ++nt) {
#pragma unroll
        for (int v = 0; v < 8; ++v) {
            int r = tile * 16 + v + 8 * hi;
            out[(size_t)r * D + nt * 16 + col] = accS[nt][v] * sc[v] + accT[nt][v];
        }
    }
}

// ----------------------------------------------------------------- batch norm
__global__ __launch_bounds__(256) void bn_stats_kernel(const float* __restrict__ h,
                                                       float* __restrict__ stats) {
    __shared__ float red[2][256];
    int c = threadIdx.x & 63;
    int g = threadIdx.x >> 6;
    int base = blockIdx.x * 512;
    float s = 0.0f, q = 0.0f;
    for (int j = 0; j < 128; ++j) {
        int r = base + g + j * 4;
        if (r < N_NODES) {
            float v = h[(size_t)r * D + c];
            s += v;
            q += v * v;
        }
    }
    red[0][threadIdx.x] = s;
    red[1][threadIdx.x] = q;
    __syncthreads();
    if (g == 0) {
        s = red[0][c] + red[0][64 + c] + red[0][128 + c] + red[0][192 + c];
        q = red[1][c] + red[1][64 + c] + red[1][128 + c] + red[1][192 + c];
        atomicAdd(&stats[c], s);
        atomicAdd(&stats[64 + c], q);
    }
}

__global__ void bn_finalize_kernel(const float* __restrict__ stats, float* __restrict__ bnp) {
    int c = threadIdx.x;
    if (c < 64) {
        float mu = stats[c] / (float)N_NODES;
        float var = stats[64 + c] / (float)N_NODES - mu * mu;   // biased, like jnp.var
        bnp[c] = mu;
        bnp[64 + c] = rsqrtf(var + BN_EPS);
    }
}

__global__ void bn_relu_kernel(float* __restrict__ h, float* __restrict__ xloc,
                               const float* __restrict__ bnp,
                               const float* __restrict__ gamma,
                               const float* __restrict__ beta) {
    size_t idx = ((size_t)blockIdx.x * blockDim.x + threadIdx.x) * 4;
    if (idx >= (size_t)N_NODES * D) return;
    int c = (int)(idx & 63);
    float4 v = *(const float4*)(h + idx);
    float o[4] = {v.x, v.y, v.z, v.w};
#pragma unroll
    for (int j = 0; j < 4; ++j) {
        int ch = c + j;
        float t = (o[j] - bnp[ch]) * bnp[64 + ch] * gamma[ch] + beta[ch];
        o[j] = fmaxf(t, 0.0f);
    }
    float4 r = {o[0], o[1], o[2], o[3]};
    *(float4*)(h + idx) = r;
    float4 x0 = *(const float4*)(xloc + idx);
    x0.x += r.x; x0.y += r.y; x0.z += r.z; x0.w += r.w;
    *(float4*)(xloc + idx) = x0;
}

// --------------------------------------------------------------- residual VQ
__global__ void cbnorm_kernel(const float* __restrict__ cb, float* __restrict__ cbn) {
    int rowid = blockIdx.x * blockDim.x + threadIdx.x;
    if (rowid >= L * RVQ * CODES) return;
    const float* p = cb + (size_t)rowid * D;
    float s = 0.0f;
    for (int d = 0; d < D; ++d) s += p[d] * p[d];
    float sc = 1.0f / (sqrtf(s) + 1e-12f);
    for (int d = 0; d < D; ++d) cbn[(size_t)rowid * D + d] = p[d] * sc;
}

__global__ __launch_bounds__(256) void rvq_kernel(const float* __restrict__ h,
                                                  const float* __restrict__ cbn_layer,
                                                  float* __restrict__ ids_out,
                                                  float* __restrict__ lossacc, int layer) {
    __shared__ float scb[RVQ * CODES * D];   // 12 KB
    for (int t = threadIdx.x; t < RVQ * CODES * D; t += 256) scb[t] = cbn_layer[t];
    __syncthreads();

    int n = blockIdx.x * blockDim.x + threadIdx.x;
    float lloss = 0.0f;
    if (n < N_NODES) {
        float r[D];
        const float* hp = h + (size_t)n * D;
#pragma unroll
        for (int d = 0; d < D; ++d) r[d] = hp[d];
#pragma unroll
        for (int lv = 0; lv < RVQ; ++lv) {
            const float* cb = scb + lv * CODES * D;
            // argmax of cosine sim == argmax of raw dot (positive per-node scale)
            float best = -3.4e38f;
            int bi = 0;
            for (int c = 0; c < CODES; ++c) {
                float dot = 0.0f;
#pragma unroll
                for (int d = 0; d < D; ++d) dot += r[d] * cb[c * D + d];
                if (dot > best) { best = dot; bi = c; }
            }
            const float* q = cb + bi * D;
            float e = 0.0f;
#pragma unroll
            for (int d = 0; d < D; ++d) {
                float df = q[d] - r[d];
                e += df * df;
                r[d] -= q[d];
            }
            lloss += e;
            ids_out[(size_t)n * (L * RVQ) + layer * RVQ + lv] = (float)bi;
        }
    }
    __shared__ float lred[256];
    lred[threadIdx.x] = lloss;
    __syncthreads();
    for (int s = 128; s > 0; s >>= 1) {
        if (threadIdx.x < s) lred[threadIdx.x] += lred[threadIdx.x + s];
        __syncthreads();
    }
    if (threadIdx.x == 0) atomicAdd(lossacc, lred[0]);
}

// ---------------------------------------------------------------------- heads
__global__ __launch_bounds__(256) void pred_kernel(const float* __restrict__ x,
                                                   const float* __restrict__ Wp,
                                                   const float* __restrict__ bp,
                                                   float* __restrict__ out) {
    __shared__ float sW[D * D];
    for (int t = threadIdx.x; t < D * D; t += 256) {
        int k = t >> 6, n = t & 63;
        sW[n * 64 + k] = Wp[t];
    }
    __syncthreads();
    int wave = threadIdx.x >> 5, lane = threadIdx.x & 31;
    int tile = blockIdx.x * 8 + wave;
    if (tile >= NTILES) return;
    int hi = lane >> 4, off = hi * 2, col = lane & 15;
    int row = tile * 16 + col;
    v2f a[16];
    const float* xr = x + (size_t)row * D;
#pragma unroll
    for (int kk = 0; kk < 16; ++kk) a[kk] = *(const v2f*)(xr + kk * 4 + off);
#pragma unroll
    for (int nt = 0; nt < 4; ++nt) {
        v8f acc = v8f_zero();
        const float* pw = sW + (nt * 16 + col) * 64 + off;
#pragma unroll
        for (int kk = 0; kk < 16; ++kk)
            acc = wmma_f32(a[kk], *(const v2f*)(pw + kk * 4), acc);
        float b = bp[nt * 16 + col];
#pragma unroll
        for (int v = 0; v < 8; ++v)
            out[(size_t)(tile * 16 + v + 8 * hi) * D + nt * 16 + col] = acc[v] + b;
    }
}

__global__ __launch_bounds__(256) void gnnid_kernel(const float* __restrict__ x,
                                                    const float* __restrict__ Wg,
                                                    const float* __restrict__ bg,
                                                    float* __restrict__ out) {
    __shared__ float sW[D * 9];
    __shared__ float sb[9];
    for (int t = threadIdx.x; t < D * 9; t += 256) sW[t] = Wg[t];
    if (threadIdx.x < 9) sb[threadIdx.x] = bg[threadIdx.x];
    __syncthreads();
    int n = blockIdx.x * blockDim.x + threadIdx.x;
    if (n >= N_NODES) return;
    const float* xr = x + (size_t)n * D;
    float acc[9];
#pragma unroll
    for (int j = 0; j < 9; ++j) acc[j] = sb[j];
    for (int d = 0; d < D; ++d) {
        float v = xr[d];
#pragma unroll
        for (int j = 0; j < 9; ++j) acc[j] += v * sW[d * 9 + j];
    }
#pragma unroll
    for (int j = 0; j < 9; ++j) out[(size_t)n * 9 + j] = acc[j];
}

__global__ void loss_kernel(const float* __restrict__ lossacc, float* __restrict__ out) {
    if (threadIdx.x == 0)
        out[0] = COMMIT_W * lossacc[0] / ((float)N_NODES * (float)D);
}

// ------------------------------------------------------------------- launcher
extern "C" void kernel_launch(void* const* d_in, const int* in_sizes, int n_in,
                              void* d_out, int out_size, void* d_ws, size_t ws_size,
                              hipStream_t stream) {
    const float* x    = (const float*)d_in[0];
    const int*   ei   = (const int*)d_in[1];
    const float* Wa   = (const float*)d_in[2];
    const float* ba   = (const float*)d_in[3];
    const float* Wr   = (const float*)d_in[4];
    const float* Wl   = (const float*)d_in[5];
    const float* bl   = (const float*)d_in[6];
    const float* gamma= (const float*)d_in[7];
    const float* beta = (const float*)d_in[8];
    const float* cb   = (const float*)d_in[9];
    const float* Wp   = (const float*)d_in[10];
    const float* bp   = (const float*)d_in[11];
    const float* Wg   = (const float*)d_in[12];
    const float* bg   = (const float*)d_in[13];

    const size_t NF = (size_t)N_NODES * D;
    float* ws   = (float*)d_ws;
    float* h1   = ws;
    float* h2   = h1 + NF;
    float* agg  = h2 + NF;
    float* xloc = agg + NF;
    float* cnt  = xloc + NF;
    float* stats   = cnt + N_NODES;          // 128
    float* bnp     = stats + 128;            // 128
    float* cbn     = bnp + 128;              // L*RVQ*CODES*D = 9216
    float* lossacc = cbn + L * RVQ * CODES * D;

    float* out     = (float*)d_out;
    float* pred    = out;                    // [N, 64]
    float* commit  = out + NF;               // [1]
    float* ids     = commit + 1;             // [N, 9]
    float* gnn     = ids + (size_t)N_NODES * L * RVQ;   // [N, 9]

    const int nfBlocks   = (int)(NF / 256);             // 25000
    const int gemmBlocks = (NTILES + 7) / 8;            // 782
    const int nodeBlocks = (N_NODES + 255) / 256;       // 391

    zero_kernel<<<nfBlocks, 256, 0, stream>>>(xloc, (int)NF);
    zero_kernel<<<1, 256, 0, stream>>>(lossacc, 1);
    cbnorm_kernel<<<1, 256, 0, stream>>>(cb, cbn);

    const float* hcur = x;
    float* bufs[2] = {h1, h2};
    for (int i = 0; i < L; ++i) {
        float* hnext = bufs[i & 1];
        zero_kernel<<<nfBlocks, 256, 0, stream>>>(agg, (int)NF);
        zero_kernel<<<(N_NODES + 255) / 256, 256, 0, stream>>>(cnt, N_NODES);
        zero_kernel<<<1, 256, 0, stream>>>(stats, 128);

        edge_cnt_kernel<<<(N_EDGES + 255) / 256, 256, 0, stream>>>(ei, cnt);
        edge_agg_kernel<<<(N_EDGES * 16) / 256, 256, 0, stream>>>(ei, hcur, agg);

        sage_gemm_kernel<<<gemmBlocks, 256, 0, stream>>>(
            hcur, agg, cnt,
            Wa + (size_t)i * D * D, ba + (size_t)i * D,
            Wr + (size_t)i * D * D,
            Wl + (size_t)i * D * D, bl + (size_t)i * D,
            hnext);

        bn_stats_kernel<<<(N_NODES + 511) / 512, 256, 0, stream>>>(hnext, stats);
        bn_finalize_kernel<<<1, 64, 0, stream>>>(stats, bnp);
        bn_relu_kernel<<<(int)(NF / 4 / 256), 256, 0, stream>>>(hnext, xloc, bnp,
                                                                gamma + (size_t)i * D,
                                                                beta + (size_t)i * D);

        rvq_kernel<<<nodeBlocks, 256, 0, stream>>>(hnext, cbn + (size_t)i * RVQ * CODES * D,
                                                   ids, lossacc, i);
        hcur = hnext;
    }

    pred_kernel<<<gemmBlocks, 256, 0, stream>>>(xloc, Wp, bp, pred);
    gnnid_kernel<<<nodeBlocks, 256, 0, stream>>>(xloc, Wg, bg, gnn);
    loss_kernel<<<1, 32, 0, stream>>>(lossacc, commit);
}